// Self_L1_53927609369064
// MI455X (gfx1250) — compile-verified
//
#include <hip/hip_runtime.h>
#include <cmath>

#define N_ROWS 128
#define DIM    4096
#define KC     16                 // K per staged chunk (per-wave K slice = 1024)
#define KPAD   20                 // padded LDS row stride: 20r+2h distinct mod 64 -> conflict-free
#define NCHUNK ((DIM / 4) / KC)   // 64 chunks per wave

typedef __attribute__((ext_vector_type(2))) float v2f;
typedef __attribute__((ext_vector_type(8))) float v8f;

// Inputs are in [1e-3, 1]: raw v_log_f32 (log2) * ln(2). TRANS op co-executes with WMMA.
__device__ __forceinline__ float flog(float x) {
  return __builtin_amdgcn_logf(x) * 0.69314718055994530942f;
}

// One lane: 16B global -> LDS, tracked on ASYNCcnt (no VGPR round trip).
__device__ __forceinline__ void async_b128(const float* g, const float* l) {
  asm volatile("global_load_async_to_lds_b128 %0, %1, off"
               :: "v"((unsigned)(unsigned long long)l), "v"(g)
               : "memory");
}

// ---------------------------------------------------------------------------
// Kernel 0: ws[1+i] = sum_d ps*log(ps) - pt*log(pt);  ws[0] = 0
// ---------------------------------------------------------------------------
__global__ __launch_bounds__(256) void kl_self_kernel(const float* __restrict__ ps,
                                                      const float* __restrict__ pt,
                                                      float* __restrict__ ws) {
  const int i = blockIdx.x;                       // row 0..127
  const float* rs = ps + (size_t)i * DIM;
  const float* rt = pt + (size_t)i * DIM;
  float acc = 0.f;
  for (int c = threadIdx.x * 4; c < DIM; c += 256 * 4) {
    float4 a = *(const float4*)(rs + c);
    float4 b = *(const float4*)(rt + c);
    acc += a.x * flog(a.x) + a.y * flog(a.y) + a.z * flog(a.z) + a.w * flog(a.w);
    acc -= b.x * flog(b.x) + b.y * flog(b.y) + b.z * flog(b.z) + b.w * flog(b.w);
  }
  #pragma unroll
  for (int off = 16; off > 0; off >>= 1) acc += __shfl_down(acc, off, 32);
  __shared__ float red[8];
  const int lane = threadIdx.x & 31, wave = threadIdx.x >> 5;
  if (lane == 0) red[wave] = acc;
  __syncthreads();
  if (threadIdx.x == 0) {
    float s = 0.f;
    #pragma unroll
    for (int w = 0; w < 8; ++w) s += red[w];
    ws[1 + i] = s;
    if (i == 0) ws[0] = 0.f;
  }
}

// ---------------------------------------------------------------------------
// Kernel 1: one 16x16 output tile per block; 4 waves split K (1024 each).
// cross[i,j] = sum_d p[i,d] * log(p[j,d]) via V_WMMA_F32_16X16X4_F32.
// Staging: async global->LDS, double buffered, s_wait_asynccnt pipelined.
// ---------------------------------------------------------------------------
__global__ __launch_bounds__(128) void kl_cross_kernel(const float* __restrict__ ps,
                                                       const float* __restrict__ pt,
                                                       float* __restrict__ ws) {
  __shared__ float lds[4][2][4][16][KPAD];        // [wave][buf][As,Bs,At,Bt][row][col] = 40KB
  __shared__ float tile[4][16][16];
  __shared__ float red[4];

  const int tid  = threadIdx.x;
  const int wave = tid >> 5, lane = tid & 31;
  const int h = lane >> 4, r = lane & 15;
  const int i0 = (blockIdx.x >> 3) * 16;
  const int j0 = (blockIdx.x & 7) * 16;
  const int kBase = wave * (DIM / 4);

  constexpr int MATS = 16 * KPAD;                 // matrix stride in floats
  constexpr int BUFS = 4 * MATS;                  // buffer stride in floats
  float* myLds = &lds[wave][0][0][0][0];

  // per-lane staging coords: 2 float4 per matrix buffer (16 rows x 4 float4)
  const int sRow0 = lane >> 2,        sCol0 = (lane & 3) << 2;
  const int sRow1 = (lane + 32) >> 2, sCol1 = (lane & 3) << 2;

  auto stage = [&](int b, int k0) {
    float* base = myLds + b * BUFS;
    const float* gAs = ps + (size_t)i0 * DIM + k0;
    const float* gBs = ps + (size_t)j0 * DIM + k0;
    const float* gAt = pt + (size_t)i0 * DIM + k0;
    const float* gBt = pt + (size_t)j0 * DIM + k0;
    async_b128(gAs + (size_t)sRow0 * DIM + sCol0, base + 0 * MATS + sRow0 * KPAD + sCol0);
    async_b128(gBs + (size_t)sRow0 * DIM + sCol0, base + 1 * MATS + sRow0 * KPAD + sCol0);
    async_b128(gAt + (size_t)sRow0 * DIM + sCol0, base + 2 * MATS + sRow0 * KPAD + sCol0);
    async_b128(gBt + (size_t)sRow0 * DIM + sCol0, base + 3 * MATS + sRow0 * KPAD + sCol0);
    async_b128(gAs + (size_t)sRow1 * DIM + sCol1, base + 0 * MATS + sRow1 * KPAD + sCol1);
    async_b128(gBs + (size_t)sRow1 * DIM + sCol1, base + 1 * MATS + sRow1 * KPAD + sCol1);
    async_b128(gAt + (size_t)sRow1 * DIM + sCol1, base + 2 * MATS + sRow1 * KPAD + sCol1);
    async_b128(gBt + (size_t)sRow1 * DIM + sCol1, base + 3 * MATS + sRow1 * KPAD + sCol1);
  };

  v8f accS = {};
  v8f accT = {};

  auto compute = [&](int b) {
    const float* base = myLds + b * BUFS;
    #pragma unroll
    for (int t = 0; t < KC / 4; ++t) {
      const int c = 4 * t + 2 * h;
      v2f aS = { base[0 * MATS + r * KPAD + c], base[0 * MATS + r * KPAD + c + 1] };
      v2f bS = { flog(base[1 * MATS + r * KPAD + c]), flog(base[1 * MATS + r * KPAD + c + 1]) };
      v2f aT = { base[2 * MATS + r * KPAD + c], base[2 * MATS + r * KPAD + c + 1] };
      v2f bT = { flog(base[3 * MATS + r * KPAD + c]), flog(base[3 * MATS + r * KPAD + c + 1]) };
      accS = __builtin_amdgcn_wmma_f32_16x16x4_f32(false, aS, false, bS,
                                                   (short)0, accS, false, false);
      accT = __builtin_amdgcn_wmma_f32_16x16x4_f32(false, aT, false, bT,
                                                   (short)0, accT, false, false);
    }
  };

  stage(0, kBase);
  for (int c = 0; c < NCHUNK; ++c) {
    if (c + 1 < NCHUNK) {
      stage((c + 1) & 1, kBase + (c + 1) * KC);       // prefetch next chunk
      asm volatile("s_wait_asynccnt 0x8" ::: "memory"); // chunk c landed (in-order done)
    } else {
      asm volatile("s_wait_asynccnt 0x0" ::: "memory");
    }
    compute(c & 1);
  }

  // C/D layout: VGPR v at lane (h,r) is element (i_local = v + 8h, j_local = r).
  #pragma unroll
  for (int v = 0; v < 8; ++v)
    tile[wave][v + 8 * h][r] = accS[v] - accT[v];
  __syncthreads();

  float part = 0.f;
  for (int e = tid; e < 256; e += 128) {
    const int ii = e >> 4, jj = e & 15;
    const float cd = tile[0][ii][jj] + tile[1][ii][jj] + tile[2][ii][jj] + tile[3][ii][jj];
    part += fabsf(ws[1 + i0 + ii] - cd);
  }
  #pragma unroll
  for (int off = 16; off > 0; off >>= 1) part += __shfl_down(part, off, 32);
  if (lane == 0) red[wave] = part;
  __syncthreads();
  if (tid == 0) atomicAdd(ws, red[0] + red[1] + red[2] + red[3]);
}

// ---------------------------------------------------------------------------
// Kernel 2: final scalar
// ---------------------------------------------------------------------------
__global__ void kl_finalize_kernel(const float* __restrict__ ws, float* __restrict__ out) {
  if (threadIdx.x == 0 && blockIdx.x == 0)
    out[0] = (1e-4f + ws[0]) * (1.0f / ((float)N_ROWS * (float)N_ROWS));
}

extern "C" void kernel_launch(void* const* d_in, const int* in_sizes, int n_in,
                              void* d_out, int out_size, void* d_ws, size_t ws_size,
                              hipStream_t stream) {
  const float* ps = (const float*)d_in[0];   // new_source [128,4096] f32
  const float* pt = (const float*)d_in[1];   // new_target [128,4096] f32
  float* ws  = (float*)d_ws;                 // ws[0]=accum, ws[1..128]=selfdiff
  float* out = (float*)d_out;                // scalar f32

  kl_self_kernel<<<N_ROWS, 256, 0, stream>>>(ps, pt, ws);
  kl_cross_kernel<<<64, 128, 0, stream>>>(ps, pt, ws);
  kl_finalize_kernel<<<1, 64, 0, stream>>>(ws, out);
}